// TritonExpertDispatch_30554397344327
// MI455X (gfx1250) — compile-verified
//
#include <hip/hip_runtime.h>
#include <hip/hip_bf16.h>
#include <stdint.h>

#define NUM_TOKENS  16384
#define EMBED_DIM   2048
#define NUM_EXPERTS 64
#define CAPACITY    320
#define TPB         256

// d_out layout (in 4-byte elements)
#define CW_OFF   (64ULL * 320ULL * 2048ULL)            // 41,943,040
#define TI_OFF   (CW_OFF + 64ULL * 320ULL)             // 41,963,520
#define DROP_OFF (TI_OFF + 64ULL * 320ULL)             // 41,984,000

// ---------------- CDNA5 async global<->LDS helpers (ASYNCcnt path) ----------
__device__ __forceinline__ void async_load_b128(uint32_t lds_off, const void* gaddr) {
    asm volatile("global_load_async_to_lds_b128 %0, %1, off"
                 :: "v"(lds_off), "v"(gaddr) : "memory");
}
__device__ __forceinline__ void async_store_b128(const void* gaddr, uint32_t lds_off) {
    asm volatile("global_store_async_from_lds_b128 %0, %1, off"
                 :: "v"(gaddr), "v"(lds_off) : "memory");
}
__device__ __forceinline__ void wait_async0() {
    asm volatile("s_wait_asynccnt 0x0" ::: "memory");
}
__device__ __forceinline__ uint32_t lds_off32(const void* p) {
    // LDS aperture: generic shared address carries LDS byte offset in addr[31:0]
    return (uint32_t)(uintptr_t)p;
}

// ---------------- Phase 1: per-block stable ranks + per-block expert counts -
__global__ void k_phase1(const int* __restrict__ expert_ids,
                         int* __restrict__ blockHist,   // [64 blocks][64 experts]
                         int* __restrict__ localRank) { // [NUM_TOKENS]
    __shared__ int hist[8][64];
    __shared__ int waveOff[8][64];
    const int tid  = threadIdx.x;
    const int lane = tid & 31;
    const int wave = tid >> 5;
    const int t    = blockIdx.x * TPB + tid;

    for (int i = tid; i < 8 * 64; i += TPB) ((int*)hist)[i] = 0;
    __syncthreads();

    int e = expert_ids[t];
    e = e < 0 ? 0 : (e > NUM_EXPERTS - 1 ? NUM_EXPERTS - 1 : e);

    // stable within-wave rank: count lower lanes with same expert (wave32)
    int r = 0;
    for (int j = 0; j < 31; ++j) {
        int ej = __shfl(e, j, 32);
        if (j < lane && ej == e) r++;
    }
    atomicAdd(&hist[wave][e], 1);
    __syncthreads();

    if (tid < 64) {
        int run = 0;
        for (int w = 0; w < 8; ++w) {
            int c = hist[w][tid];
            waveOff[w][tid] = run;
            run += c;
        }
        blockHist[blockIdx.x * 64 + tid] = run;
    }
    __syncthreads();

    localRank[t] = waveOff[wave][e] + r;
}

// ---------------- Phase 2: exclusive prefix over blocks + exact drop count --
__global__ void k_phase2(const int* __restrict__ blockHist,
                         int* __restrict__ blockOffset, // [64 blocks][64 experts]
                         int* __restrict__ dropOut) {
    __shared__ int drop[64];
    const int e = threadIdx.x; // 64 threads
    int run = 0;
    for (int b = 0; b < 64; ++b) {
        blockOffset[b * 64 + e] = run;
        run += blockHist[b * 64 + e];
    }
    drop[e] = run > CAPACITY ? run - CAPACITY : 0;
    __syncthreads();
    if (e == 0) {
        int s = 0;
        for (int i = 0; i < 64; ++i) s += drop[i];
        *dropOut = s;
    }
}

// ---------------- Phase 3: init small outputs (weights=0, indices=-1) -------
__global__ void k_init_small(float* __restrict__ cw, int* __restrict__ ti) {
    int i = blockIdx.x * blockDim.x + threadIdx.x;
    if (i < NUM_EXPERTS * CAPACITY) { cw[i] = 0.0f; ti[i] = -1; }
}

// ---------------- Phase 4: scatter metadata (unique (e,pos) per token) ------
__global__ void k_scatter_meta(const int* __restrict__ expert_ids,
                               const float* __restrict__ weights,
                               const int* __restrict__ blockOffset,
                               const int* __restrict__ localRank,
                               float* __restrict__ cw, int* __restrict__ ti) {
    const int t = blockIdx.x * TPB + threadIdx.x;
    int e = expert_ids[t];
    e = e < 0 ? 0 : (e > NUM_EXPERTS - 1 ? NUM_EXPERTS - 1 : e);
    const int pos = blockOffset[blockIdx.x * 64 + e] + localRank[t];
    if (pos < CAPACITY) {
        cw[e * CAPACITY + pos] = weights[t];
        ti[e * CAPACITY + pos] = t;
    }
}

// ---------------- Phase 5: one block per output row; write each row once ----
__global__ void k_dispatch_copy(const float* __restrict__ x,
                                const int* __restrict__ ti,
                                float* __restrict__ dx) {
    __shared__ float buf[EMBED_DIM]; // 8 KB staging
    const int slot = blockIdx.x;                    // e*CAPACITY + p
    const int tok  = ti[slot];                      // uniform per block
    float* orow = dx + (size_t)slot * EMBED_DIM;
    const int tid = threadIdx.x;

    if (tok >= 0) {
        const float* irow = x + (size_t)tok * EMBED_DIM;
        const uint32_t l0 = lds_off32(&buf[tid * 4]);
        const uint32_t l1 = lds_off32(&buf[(tid + 256) * 4]);
        async_load_b128(l0, irow + tid * 4);
        async_load_b128(l1, irow + (tid + 256) * 4);
        wait_async0();                              // loads vs stores are unordered
        async_store_b128(orow + tid * 4, l0);
        async_store_b128(orow + (tid + 256) * 4, l1);
        // s_endpgm performs implicit wait-idle for outstanding async stores
    } else {
        const float4 z = make_float4(0.f, 0.f, 0.f, 0.f);
        *(float4*)(orow + tid * 4) = z;
        *(float4*)(orow + (tid + 256) * 4) = z;
    }
}

extern "C" void kernel_launch(void* const* d_in, const int* in_sizes, int n_in,
                              void* d_out, int out_size, void* d_ws, size_t ws_size,
                              hipStream_t stream) {
    const float* x          = (const float*)d_in[0];
    const int*   expert_ids = (const int*)d_in[1];
    const float* weights    = (const float*)d_in[2];

    float* out = (float*)d_out;
    float* dx  = out;                          // [64][320][2048]
    float* cw  = out + CW_OFF;                 // [64][320]
    int*   ti  = (int*)(out + TI_OFF);         // [64][320]
    int*   dropOut = (int*)(out + DROP_OFF);   // scalar

    int* blockHist   = (int*)d_ws;             // 64*64 ints
    int* blockOffset = blockHist + 64 * 64;    // 64*64 ints
    int* localRank   = blockOffset + 64 * 64;  // 16384 ints   (total 96 KB)

    k_phase1<<<NUM_TOKENS / TPB, TPB, 0, stream>>>(expert_ids, blockHist, localRank);
    k_phase2<<<1, 64, 0, stream>>>(blockHist, blockOffset, dropOut);
    k_init_small<<<(NUM_EXPERTS * CAPACITY + TPB - 1) / TPB, TPB, 0, stream>>>(cw, ti);
    k_scatter_meta<<<NUM_TOKENS / TPB, TPB, 0, stream>>>(expert_ids, weights,
                                                         blockOffset, localRank, cw, ti);
    k_dispatch_copy<<<NUM_EXPERTS * CAPACITY, TPB, 0, stream>>>(x, ti, dx);
}